// IntraCellularAttention_8418135900519
// MI455X (gfx1250) — compile-verified
//
#include <hip/hip_runtime.h>

// CDNA5 / gfx1250 — IntraCellularAttention
// B=1024, D_INNER=4096, D_STATE=16, D_ATTN=16, f32 throughout.
// One workgroup per batch element. h[b] staged into LDS by the Tensor Data
// Mover (tensor_load_to_lds with LDS row padding), all GEMMs on
// v_wmma_f32_16x16x4_f32 with dual accumulator chains, cross-wave tile
// reduction via ds_add_f32.

typedef __attribute__((ext_vector_type(2))) float        v2f;
typedef __attribute__((ext_vector_type(8))) float        v8f;
typedef __attribute__((ext_vector_type(4))) unsigned int v4u;
typedef __attribute__((ext_vector_type(8))) int          v8i;
typedef __attribute__((ext_vector_type(4))) int          v4i;

#define BATCH    1024
#define D_INNER  4096
#define D_STATE  16
#define SCALE_F  0.25f          // D_ATTN^-0.5

#define ROWS     18             // padded LDS row stride in floats (bank-conflict-free)
#define HL_FL    (D_INNER * ROWS)       // 73728 floats
#define Q_OFF    HL_FL
#define K_OFF    (Q_OFF + 256)
#define A_OFF    (K_OFF + 256)
#define SMEM_FL  (A_OFF + 256)          // 74496 floats
#define SMEM_BYTES (SMEM_FL * 4)        // 297,984 B  (< 320 KB WGP LDS)

#define ZERO8 {0.f,0.f,0.f,0.f,0.f,0.f,0.f,0.f}

__global__ __launch_bounds__(256)
void ica_kernel(const float* __restrict__ h,
                const float* __restrict__ Wq,
                const float* __restrict__ Wk,
                const float* __restrict__ gate,
                float* __restrict__ out) {
    extern __shared__ float sm[];
    float* hL = sm;            // h[b] as 4096 rows x 16 floats, row stride 18
    float* qT = sm + Q_OFF;    // 16x16 q tile
    float* kT = sm + K_OFF;    // 16x16 k tile
    float* aT = sm + A_OFF;    // 16x16 attn tile

    const int b    = blockIdx.x;
    const int tid  = threadIdx.x;
    const int lane = tid & 31;
    const int wave = tid >> 5;
    const float g  = gate[0];

    const size_t bOfs = (size_t)b * (D_INNER * D_STATE);

    // ---------------- Stage h[b] -> LDS via Tensor Data Mover ----------------
    // One TDM descriptor moves the full 4096x16 f32 tile; pad_enable inserts
    // 2 DWORDs after every 16 DWORDs -> LDS row stride of 18 floats.
    qT[tid] = 0.0f;            // zero q/k reduction tiles meanwhile
    kT[tid] = 0.0f;
    if (wave == 0) {
        unsigned long long ga = (unsigned long long)(h + bOfs);
        unsigned ldsAddr = (unsigned)(unsigned long long)(void*)hL; // AS(3) byte offset

        v4u g0;
        g0.x = 1u;                                       // count=1 (valid user D#)
        g0.y = ldsAddr;                                  // lds_addr
        g0.z = (unsigned)(ga & 0xFFFFFFFFu);             // global_addr[31:0]
        g0.w = (unsigned)((ga >> 32) & 0x01FFFFFFu)      // global_addr[56:32]
             | (2u << 30);                               // type=2 ("image")

        v8i g1;
        g1[0] = (int)0x02D20000;   // data_size=4B | pad_enable | pad_interval=16dw | pad_amount=2dw
        g1[1] = (int)(16u << 16);  // tensor_dim0 = 16 (low half in [63:48])
        g1[2] = (int)(4096u << 16);// tensor_dim1 = 4096 (low half in [95:80])
        g1[3] = (int)(16u << 16);  // tile_dim0 = 16
        g1[4] = (int)4096;         // tile_dim1 = 4096
        g1[5] = (int)16;           // tensor_dim0_stride = 16 elements
        g1[6] = 0;
        g1[7] = 0;

        v4i gz4 = {0, 0, 0, 0};            // 2-D tile: groups 2/3 unused
        v8i gz8 = {0, 0, 0, 0, 0, 0, 0, 0};
        __builtin_amdgcn_tensor_load_to_lds(g0, g1, gz4, gz4, gz8, 0);
        __builtin_amdgcn_s_wait_tensorcnt((short)0);     // TENSORcnt -> 0
    }
    __syncthreads();

    // ---------------- Phase 1: q = slots * Wq^T, k = slots * Wk^T ----------------
    // 16x16 outputs, K=4096 split across 8 waves (512 each); K=4 per WMMA,
    // two independent accumulator chains per output to hide WMMA latency.
    {
        const int row = lane & 15;           // A: M row (s) ; B: N col (a)
        const int k0  = (lane >> 4) * 2;     // K offset per lane half
        v8f accQ0 = ZERO8, accQ1 = ZERO8, accK0 = ZERO8, accK1 = ZERO8;
        const float* wqBase = Wq + row * D_INNER + k0;
        const float* wkBase = Wk + row * D_INNER + k0;
        const int dEnd = wave * 512 + 512;
        for (int d0 = wave * 512; d0 < dEnd; d0 += 8) {
            v2f a0, a1;
            a0.x = hL[(d0 + k0)     * ROWS + row];
            a0.y = hL[(d0 + k0 + 1) * ROWS + row];
            a1.x = hL[(d0 + 4 + k0)     * ROWS + row];
            a1.y = hL[(d0 + 4 + k0 + 1) * ROWS + row];
            v2f bq0 = *(const v2f*)(wqBase + d0);
            v2f bk0 = *(const v2f*)(wkBase + d0);
            v2f bq1 = *(const v2f*)(wqBase + d0 + 4);
            v2f bk1 = *(const v2f*)(wkBase + d0 + 4);
            accQ0 = __builtin_amdgcn_wmma_f32_16x16x4_f32(false, a0, false, bq0,
                                                          (short)0, accQ0, false, false);
            accK0 = __builtin_amdgcn_wmma_f32_16x16x4_f32(false, a0, false, bk0,
                                                          (short)0, accK0, false, false);
            accQ1 = __builtin_amdgcn_wmma_f32_16x16x4_f32(false, a1, false, bq1,
                                                          (short)0, accQ1, false, false);
            accK1 = __builtin_amdgcn_wmma_f32_16x16x4_f32(false, a1, false, bk1,
                                                          (short)0, accK1, false, false);
        }
        v8f accQ = accQ0 + accQ1;
        v8f accK = accK0 + accK1;
        // Cross-wave reduction of partial 16x16 tiles via LDS float atomics.
        const int hi = lane >> 4, n = lane & 15;
        #pragma unroll
        for (int i = 0; i < 8; ++i) {
            int m = i + 8 * hi;              // C/D layout: VGPR i -> M = i + 8*(lane>=16)
            atomicAdd(&qT[m * 16 + n], accQ[i]);
            atomicAdd(&kT[m * 16 + n], accK[i]);
        }
    }
    __syncthreads();

    // ---------------- Phase 2: attn = softmax(q k^T * scale)  (tiny, scalar) ----------------
    {
        int s = tid >> 4, t = tid & 15;
        float dot = 0.f;
        #pragma unroll
        for (int a = 0; a < 16; ++a) dot += qT[s * 16 + a] * kT[t * 16 + a];
        aT[s * 16 + t] = dot * SCALE_F;
    }
    __syncthreads();
    if (tid < 16) {
        float mx = -3.402823466e38f;
        #pragma unroll
        for (int t = 0; t < 16; ++t) mx = fmaxf(mx, aT[tid * 16 + t]);
        float sum = 0.f;
        #pragma unroll
        for (int t = 0; t < 16; ++t) {
            float e = __expf(aT[tid * 16 + t] - mx);
            aT[tid * 16 + t] = e;
            sum += e;
        }
        float inv = 1.0f / sum;
        #pragma unroll
        for (int t = 0; t < 16; ++t) aT[tid * 16 + t] *= inv;
    }
    __syncthreads();

    // ---------------- Phase 3: out = attn * slots, fused gated residual ----------------
    // out is 16x4096; each wave produces 32 chunks of 16 columns.
    // K=16 as 4 WMMAs, split over two independent accumulator chains.
    {
        const int n  = lane & 15;            // A: M row ; B/D: N col
        const int hi = lane >> 4;
        const int k0 = hi * 2;
        // A fragments (attn rows) are chunk-invariant: preload all 4 (t0 = 0,4,8,12).
        v2f afr[4];
        #pragma unroll
        for (int t = 0; t < 4; ++t) {
            afr[t].x = aT[n * 16 + t * 4 + k0];
            afr[t].y = aT[n * 16 + t * 4 + k0 + 1];
        }
        float* dstBase = out + bOfs;
        for (int cc = 0; cc < 32; ++cc) {
            const int d0 = (wave * 32 + cc) * 16;
            const float* brow = hL + (d0 + n) * ROWS + k0;
            v2f b0 = *(const v2f*)(brow);
            v2f b1 = *(const v2f*)(brow + 4);
            v2f b2 = *(const v2f*)(brow + 8);
            v2f b3 = *(const v2f*)(brow + 12);
            v8f acc0 = ZERO8, acc1 = ZERO8;
            acc0 = __builtin_amdgcn_wmma_f32_16x16x4_f32(false, afr[0], false, b0,
                                                         (short)0, acc0, false, false);
            acc1 = __builtin_amdgcn_wmma_f32_16x16x4_f32(false, afr[1], false, b1,
                                                         (short)0, acc1, false, false);
            acc0 = __builtin_amdgcn_wmma_f32_16x16x4_f32(false, afr[2], false, b2,
                                                         (short)0, acc0, false, false);
            acc1 = __builtin_amdgcn_wmma_f32_16x16x4_f32(false, afr[3], false, b3,
                                                         (short)0, acc1, false, false);
            v8f acc = acc0 + acc1;
            // Writeback: lane owns column d = d0+n, rows s = 8*hi .. 8*hi+7 (contiguous).
            const int d = d0 + n;
            const int sB = 8 * hi;
            const float* res = hL + d * ROWS + sB;
            float4 o0, o1;
            o0.x = res[0] + g * acc[0];
            o0.y = res[1] + g * acc[1];
            o0.z = res[2] + g * acc[2];
            o0.w = res[3] + g * acc[3];
            o1.x = res[4] + g * acc[4];
            o1.y = res[5] + g * acc[5];
            o1.z = res[6] + g * acc[6];
            o1.w = res[7] + g * acc[7];
            float4* op = (float4*)(dstBase + d * 16 + sB);   // 16B-aligned
            op[0] = o0;
            op[1] = o1;
        }
    }
}

extern "C" void kernel_launch(void* const* d_in, const int* in_sizes, int n_in,
                              void* d_out, int out_size, void* d_ws, size_t ws_size,
                              hipStream_t stream) {
    (void)in_sizes; (void)n_in; (void)out_size; (void)d_ws; (void)ws_size;
    const float* h    = (const float*)d_in[0];
    const float* Wq   = (const float*)d_in[1];
    const float* Wk   = (const float*)d_in[2];
    const float* gate = (const float*)d_in[3];
    float* out = (float*)d_out;

    // Allow > default dynamic LDS (297,984 B; CDNA5 WGP has 320 KB).
    (void)hipFuncSetAttribute((const void*)ica_kernel,
                              hipFuncAttributeMaxDynamicSharedMemorySize, SMEM_BYTES);

    ica_kernel<<<BATCH, 256, SMEM_BYTES, stream>>>(h, Wq, Wk, gate, out);
}